// GCNLayer_86852828659807
// MI455X (gfx1250) — compile-verified
//
#include <hip/hip_runtime.h>

typedef __attribute__((ext_vector_type(2))) float v2f;
typedef __attribute__((ext_vector_type(8))) float v8f;

#define D 64
#define ALPHA 0.01f
#define BN_EPS 1e-5f

// ---------------- degree / normalization ----------------
__global__ void deg_init_k(float* __restrict__ deg, int n) {
    int i = blockIdx.x * blockDim.x + threadIdx.x;
    if (i < n) deg[i] = 1.0f;                       // self-loop weight
}

__global__ void deg_acc_k(const int* __restrict__ col, const float* __restrict__ w,
                          float* __restrict__ deg, int e) {
    int i = blockIdx.x * blockDim.x + threadIdx.x;
    if (i < e) atomicAdd(&deg[col[i]], w[i]);
}

__global__ void dis_k(const float* __restrict__ deg, float* __restrict__ dis, int n) {
    int i = blockIdx.x * blockDim.x + threadIdx.x;
    if (i < n) {
        float dg = deg[i];
        dis[i] = dg > 0.0f ? rsqrtf(dg) : 0.0f;
    }
}

// ---------------- h = x @ W via V_WMMA_F32_16X16X4_F32 ----------------
// One wave32 computes a 16x64 stripe of h. 4 accumulators (16x16 each),
// 16 K-steps of K=4. FP32 throughout for exactness vs the FP32 reference.
__global__ void gemm_xw_wmma(const float* __restrict__ x, const float* __restrict__ W,
                             float* __restrict__ h, int n) {
    int wave = (int)((blockIdx.x * blockDim.x + threadIdx.x) >> 5);
    int lane = threadIdx.x & 31;
    int tiles = n >> 4;
    if (wave >= tiles) return;                     // wave-uniform: EXEC stays all-1s

    int rowBase = wave << 4;
    int half = lane >> 4;                          // 0: K%4 in {0,1}, 1: K%4 in {2,3}
    int lid  = lane & 15;

    const float* xrow = x + (size_t)(rowBase + lid) * D;

    v8f acc0 = {}, acc1 = {}, acc2 = {}, acc3 = {};

#pragma unroll
    for (int s = 0; s < 16; ++s) {
        int k = 4 * s + 2 * half;
        // A fragment: VGPR0 = K (k), VGPR1 = K (k+1), row = rowBase+lid
        v2f a;
        a.x = xrow[k];
        a.y = xrow[k + 1];
        // B fragments: VGPR0 = row k of W, VGPR1 = row k+1; col = colBase+lid
        const float* w0 = W + (size_t)k * D;
        const float* w1 = W + (size_t)(k + 1) * D;
        v2f b0; b0.x = w0[lid +  0]; b0.y = w1[lid +  0];
        v2f b1; b1.x = w0[lid + 16]; b1.y = w1[lid + 16];
        v2f b2; b2.x = w0[lid + 32]; b2.y = w1[lid + 32];
        v2f b3; b3.x = w0[lid + 48]; b3.y = w1[lid + 48];

        acc0 = __builtin_amdgcn_wmma_f32_16x16x4_f32(false, a, false, b0, (short)0, acc0, false, false);
        acc1 = __builtin_amdgcn_wmma_f32_16x16x4_f32(false, a, false, b1, (short)0, acc1, false, false);
        acc2 = __builtin_amdgcn_wmma_f32_16x16x4_f32(false, a, false, b2, (short)0, acc2, false, false);
        acc3 = __builtin_amdgcn_wmma_f32_16x16x4_f32(false, a, false, b3, (short)0, acc3, false, false);
    }

    // D layout: VGPR r -> row rowBase + r + 8*half, col = colTile*16 + lid
#pragma unroll
    for (int r = 0; r < 8; ++r) {
        int row = rowBase + r + 8 * half;
        float* hp = h + (size_t)row * D + lid;
        hp[0]  = acc0[r];
        hp[16] = acc1[r];
        hp[32] = acc2[r];
        hp[48] = acc3[r];
    }
}

// ---------------- aggregation ----------------
__global__ void agg_init_k(const float* __restrict__ h, const float* __restrict__ dis,
                           const float* __restrict__ b, float* __restrict__ out, int total) {
    int t = blockIdx.x * blockDim.x + threadIdx.x;
    if (t >= total) return;
    int i = t >> 6, d = t & 63;
    float di = dis[i];
    out[t] = b[d] + di * di * h[t];                // self-loop message + bias
}

__global__ void edge_norm_k(const int* __restrict__ row, const int* __restrict__ col,
                            const float* __restrict__ w, const float* __restrict__ dis,
                            float* __restrict__ nrm, int e) {
    int i = blockIdx.x * blockDim.x + threadIdx.x;
    if (i < e) nrm[i] = dis[row[i]] * w[i] * dis[col[i]];
}

__global__ void scatter_k(const int* __restrict__ row, const int* __restrict__ col,
                          const float* __restrict__ nrm, const float* __restrict__ h,
                          float* __restrict__ out, int total) {
    int t = blockIdx.x * blockDim.x + threadIdx.x;
    if (t >= total) return;
    int e = t >> 6, d = t & 63;
    int r = row[e], c = col[e];
    atomicAdd(&out[(size_t)c * D + d], nrm[e] * h[(size_t)r * D + d]);
}

// ---------------- batch norm ----------------
__global__ void zero_sums_k(float* __restrict__ sums) {
    int i = threadIdx.x;
    if (i < 128) sums[i] = 0.0f;
}

__global__ void bn_stats_k(const float* __restrict__ agg, float* __restrict__ sums, int n) {
    __shared__ float ls[256];
    __shared__ float ls2[256];
    int d = threadIdx.x & 63;
    int g = threadIdx.x >> 6;                      // 4 row-groups per block
    float s = 0.0f, s2 = 0.0f;
    for (int i = blockIdx.x * 4 + g; i < n; i += gridDim.x * 4) {
        float v = agg[(size_t)i * D + d];
        s += v;
        s2 += v * v;
    }
    ls[threadIdx.x] = s;
    ls2[threadIdx.x] = s2;
    __syncthreads();
    if (g == 0) {
        s  = ls[d]  + ls[64 + d]  + ls[128 + d]  + ls[192 + d];
        s2 = ls2[d] + ls2[64 + d] + ls2[128 + d] + ls2[192 + d];
        atomicAdd(&sums[d], s);
        atomicAdd(&sums[64 + d], s2);
    }
}

__global__ void bn_finalize_k(const float* __restrict__ sums, const float* __restrict__ gamma,
                              const float* __restrict__ beta, float* __restrict__ ss, float inv_n) {
    int d = threadIdx.x;
    if (d >= 64) return;
    float mean = sums[d] * inv_n;
    float var  = sums[64 + d] * inv_n - mean * mean;
    float sc   = gamma[d] * rsqrtf(var + BN_EPS);
    ss[d]      = sc;
    ss[64 + d] = beta[d] - mean * sc;
}

__global__ void bn_apply_k(float* __restrict__ out, const float* __restrict__ ss, int total) {
    int t = blockIdx.x * blockDim.x + threadIdx.x;
    if (t >= total) return;
    int d = t & 63;
    float v = out[t] * ss[d] + ss[64 + d];
    out[t] = v >= 0.0f ? v : ALPHA * v;
}

// ---------------- launcher ----------------
static inline size_t align256(size_t v) { return (v + 255) & ~(size_t)255; }

extern "C" void kernel_launch(void* const* d_in, const int* in_sizes, int n_in,
                              void* d_out, int out_size, void* d_ws, size_t ws_size,
                              hipStream_t stream) {
    const float* x     = (const float*)d_in[0];
    const int*   eidx  = (const int*)d_in[1];
    const float* eattr = (const float*)d_in[2];
    const float* W     = (const float*)d_in[3];
    const float* bias  = (const float*)d_in[4];
    const float* gamma = (const float*)d_in[5];
    const float* beta  = (const float*)d_in[6];
    float* out = (float*)d_out;

    const int N = in_sizes[0] / D;                 // 100000
    const int E = in_sizes[2];                     // 1200000
    const int* row = eidx;                         // edge_idx[0, :]
    const int* col = eidx + E;                     // edge_idx[1, :]

    char* ws = (char*)d_ws;
    float* deg  = (float*)ws;  ws += align256((size_t)N * sizeof(float));
    float* dis  = (float*)ws;  ws += align256((size_t)N * sizeof(float));
    float* h    = (float*)ws;  ws += align256((size_t)N * D * sizeof(float));
    float* nrm  = (float*)ws;  ws += align256((size_t)E * sizeof(float));
    float* sums = (float*)ws;  ws += align256(128 * sizeof(float));
    float* ss   = (float*)ws;

    const int B = 256;
    const int totalND = N * D;                     // 6.4M
    const int totalED = E * D;                     // 76.8M (< 2^31)

    // 1) degrees + deg^-1/2
    deg_init_k<<<(N + B - 1) / B, B, 0, stream>>>(deg, N);
    deg_acc_k<<<(E + B - 1) / B, B, 0, stream>>>(col, eattr, deg, E);
    dis_k<<<(N + B - 1) / B, B, 0, stream>>>(deg, dis, N);

    // 2) h = x @ W   (WMMA f32 16x16x4, one wave per 16-row stripe)
    int tiles = N / 16;                            // 6250, exact
    int wavesPerBlock = B / 32;                    // 8
    gemm_xw_wmma<<<(tiles + wavesPerBlock - 1) / wavesPerBlock, B, 0, stream>>>(x, W, h, N);

    // 3) aggregate: self-loops + bias, then edge scatter
    agg_init_k<<<(totalND + B - 1) / B, B, 0, stream>>>(h, dis, bias, out, totalND);
    edge_norm_k<<<(E + B - 1) / B, B, 0, stream>>>(row, col, eattr, dis, nrm, E);
    scatter_k<<<(totalED + B - 1) / B, B, 0, stream>>>(row, col, nrm, h, out, totalED);

    // 4) batch norm stats + 5) fused affine + LeakyReLU
    zero_sums_k<<<1, 128, 0, stream>>>(sums);
    bn_stats_k<<<256, B, 0, stream>>>(out, sums, N);
    bn_finalize_k<<<1, 64, 0, stream>>>(sums, gamma, beta, ss, 1.0f / (float)N);
    bn_apply_k<<<(totalND + B - 1) / B, B, 0, stream>>>(out, ss, totalND);
}